// ChebConv_42941083025912
// MI455X (gfx1250) — compile-verified
//
#include <hip/hip_runtime.h>

// ---------------- problem constants (from reference) ----------------
#define V_NODES 100000
#define B_BATCH 2
#define FIN     64
#define K_CHEB  3
#define FOUT    64
#define COLS    (FIN * B_BATCH)       // 128 columns of the x0/x1/x2 planes
#define E_EDGES 1600000
#define KDIM    (FIN * K_CHEB)        // 192 = GEMM reduction dim

typedef __attribute__((ext_vector_type(2))) float v2f;
typedef __attribute__((ext_vector_type(4))) float v4f;
typedef __attribute__((ext_vector_type(8))) float v8f;

// -------------------------------------------------------------------
// Stage 1: x0[v, f*B+b] = x[b,v,f]; x1 = 0; x2 = -x0 (folds the "-x0"
// term of the Chebyshev recurrence into the init so SpMM #2 can
// accumulate 2*L*x1 directly on top).
// -------------------------------------------------------------------
__global__ void cheb_build(const float* __restrict__ x,
                           float* __restrict__ x0,
                           float* __restrict__ x1,
                           float* __restrict__ x2) {
  long long t = (long long)blockIdx.x * blockDim.x + threadIdx.x;
  const long long total = (long long)V_NODES * COLS;
  if (t >= total) return;
  int v = (int)(t / COLS);
  int c = (int)(t % COLS);
  int f = c >> 1;          // col = f*B + b, B==2
  int b = c & 1;
  float val = x[((long long)b * V_NODES + v) * FIN + f];
  x0[t] = val;
  x1[t] = 0.0f;
  x2[t] = -val;
}

// -------------------------------------------------------------------
// Stage 2/3: edge-parallel COO SpMM. One wave32 per edge; each lane
// handles 4 columns (coalesced 128B per 32-lane group). The 51MB
// src/dst planes stay resident in the 192MB L2, so the gathers and
// f32 atomics resolve at L2 bandwidth.
//   dst[rows[e], :] += scale * vals[e] * src[cols[e], :]
// -------------------------------------------------------------------
__global__ void cheb_spmm(const float* __restrict__ vals,
                          const int*   __restrict__ rows,
                          const int*   __restrict__ cols,
                          const float* __restrict__ src,
                          float*       __restrict__ dst,
                          float scale) {
  int wave = (int)(((long long)blockIdx.x * blockDim.x + threadIdx.x) >> 5);
  int lane = threadIdx.x & 31;
  if (wave >= E_EDGES) return;
  float a = vals[wave] * scale;
  long long rbase = (long long)rows[wave] * COLS;
  long long cbase = (long long)cols[wave] * COLS;
#pragma unroll
  for (int i = 0; i < COLS / 32; ++i) {
    int col = lane + i * 32;
    float contrib = a * src[cbase + col];
    __hip_atomic_fetch_add(&dst[rbase + col], contrib,
                           __ATOMIC_RELAXED, __HIP_MEMORY_SCOPE_AGENT);
  }
}

// -------------------------------------------------------------------
// Stage 4: out[m, n] = sum_c A(m,c) * W[c, n] + bias[n]
//   m = b*V + v   (200000 rows, divisible by 16)
//   c = f*K + k   (192) ;  A(m,c) = plane_k[v*128 + f*2 + b]
//   W flat = weight[f,k,o] = W[(f*3+k)*64 + n]
//
// Reduction is re-ordered plane-by-plane (k outer, f inner) so every
// WMMA k-step reads ONE plane with a wave-uniform base pointer ->
// global_load (saddr) instead of flat_load + cndmask selects.
// One wave computes a 16x64 stripe: 3 planes x 16 steps x 4 N-tiles
// = 192 V_WMMA_F32_16X16X4_F32. W (48KB) is staged in LDS per block.
// -------------------------------------------------------------------
__device__ __forceinline__ void gemm_plane(const float* __restrict__ plane,
                                           int k, const float* __restrict__ sW,
                                           long long aoff, int half, int l,
                                           v8f& acc0, v8f& acc1, v8f& acc2, v8f& acc3) {
  for (int f0 = 0; f0 < FIN; f0 += 4) {         // 16 k-steps per plane
    const int fA = f0 + half * 2;               // this half-wave's K pair
    // A fragment: 16x4 f32 (lanes 0-15: K=fA,fA+1 ; lanes 16-31: K=f0+2,f0+3)
    v2f a;
    a.x = plane[aoff + (long long)fA * B_BATCH];
    a.y = plane[aoff + (long long)(fA + 1) * B_BATCH];
    // B fragments from LDS: rows r0/r1 of W for 4 N-tiles of 16.
    const int r0 = (fA * K_CHEB + k) * FOUT;
    const int r1 = r0 + K_CHEB * FOUT;          // (fA+1)*3+k row
    v2f b0, b1, b2, b3;
    b0.x = sW[r0 + l +  0]; b0.y = sW[r1 + l +  0];
    b1.x = sW[r0 + l + 16]; b1.y = sW[r1 + l + 16];
    b2.x = sW[r0 + l + 32]; b2.y = sW[r1 + l + 32];
    b3.x = sW[r0 + l + 48]; b3.y = sW[r1 + l + 48];
    acc0 = __builtin_amdgcn_wmma_f32_16x16x4_f32(false, a, false, b0, (short)0, acc0, false, false);
    acc1 = __builtin_amdgcn_wmma_f32_16x16x4_f32(false, a, false, b1, (short)0, acc1, false, false);
    acc2 = __builtin_amdgcn_wmma_f32_16x16x4_f32(false, a, false, b2, (short)0, acc2, false, false);
    acc3 = __builtin_amdgcn_wmma_f32_16x16x4_f32(false, a, false, b3, (short)0, acc3, false, false);
  }
}

__device__ __forceinline__ void store_tile(const v8f acc, int tileM, int n,
                                           int half, float bval,
                                           float* __restrict__ out) {
#pragma unroll
  for (int r = 0; r < 8; ++r) {
    int mrow = tileM + r + half * 8;            // C/D layout: VGPR r -> M=r / r+8
    out[(long long)mrow * FOUT + n] = acc[r] + bval;
  }
}

__global__ void __launch_bounds__(256)
cheb_gemm(const float* __restrict__ x0,
          const float* __restrict__ x1,
          const float* __restrict__ x2,
          const float* __restrict__ w,     // [FIN,K,FOUT] flat
          const float* __restrict__ bias,
          float*       __restrict__ out) {
  __shared__ float sW[KDIM * FOUT];             // 48KB: whole weight matrix

  // Cooperative LDS fill (float4, fully coalesced) BEFORE any early-exit.
  {
    const v4f* __restrict__ w4 = (const v4f*)w;
    v4f* sW4 = (v4f*)sW;
    for (int i = threadIdx.x; i < (KDIM * FOUT) / 4; i += 256)
      sW4[i] = w4[i];
  }
  __syncthreads();

  const int wave = (int)(((long long)blockIdx.x * blockDim.x + threadIdx.x) >> 5);
  const int lane = threadIdx.x & 31;
  const int numTiles = (B_BATCH * V_NODES) / 16;  // 12500
  if (wave >= numTiles) return;

  const int tileM = wave * 16;
  const int half  = lane >> 4;                  // 0: K0/K1 lanes, 1: K2/K3 lanes
  const int l     = lane & 15;

  // Row this lane supplies for the A fragment.
  const int m  = tileM + l;
  const int bb = m / V_NODES;
  const int vv = m - bb * V_NODES;
  const long long aoff = (long long)vv * COLS + bb;   // plane[aoff + f*2]

  v8f acc0 = {}, acc1 = {}, acc2 = {}, acc3 = {};

  gemm_plane(x0, 0, sW, aoff, half, l, acc0, acc1, acc2, acc3);
  gemm_plane(x1, 1, sW, aoff, half, l, acc0, acc1, acc2, acc3);
  gemm_plane(x2, 2, sW, aoff, half, l, acc0, acc1, acc2, acc3);

  store_tile(acc0, tileM, l +  0, half, bias[l +  0], out);
  store_tile(acc1, tileM, l + 16, half, bias[l + 16], out);
  store_tile(acc2, tileM, l + 32, half, bias[l + 32], out);
  store_tile(acc3, tileM, l + 48, half, bias[l + 48], out);
}

// -------------------------------------------------------------------
extern "C" void kernel_launch(void* const* d_in, const int* in_sizes, int n_in,
                              void* d_out, int out_size, void* d_ws, size_t ws_size,
                              hipStream_t stream) {
  const float* x      = (const float*)d_in[0];   // [B,V,FIN]
  const float* weight = (const float*)d_in[1];   // [FIN,K,FOUT]
  const float* bias   = (const float*)d_in[2];   // [FOUT]
  const float* vals   = (const float*)d_in[3];   // [E]
  const int*   rows   = (const int*)  d_in[4];   // [E]
  const int*   cols   = (const int*)  d_in[5];   // [E]
  float*       out    = (float*)d_out;           // [B,V,FOUT]

  const long long plane = (long long)V_NODES * COLS;  // 12.8M floats = 51.2 MB
  float* x0 = (float*)d_ws;
  float* x1 = x0 + plane;
  float* x2 = x1 + plane;                              // total 153.6 MB in d_ws

  // Stage 1: transpose + init (12.8M threads)
  {
    int threads = 256;
    int blocks  = (int)((plane + threads - 1) / threads);   // 50000
    cheb_build<<<blocks, threads, 0, stream>>>(x, x0, x1, x2);
  }
  // Stage 2: x1 = L @ x0   (one wave per edge)
  {
    int threads = 256;
    long long totalThreads = (long long)E_EDGES * 32;
    int blocks = (int)((totalThreads + threads - 1) / threads); // 200000
    cheb_spmm<<<blocks, threads, 0, stream>>>(vals, rows, cols, x0, x1, 1.0f);
  }
  // Stage 3: x2 += 2 * (L @ x1)   (x2 pre-initialized to -x0)
  {
    int threads = 256;
    long long totalThreads = (long long)E_EDGES * 32;
    int blocks = (int)((totalThreads + threads - 1) / threads);
    cheb_spmm<<<blocks, threads, 0, stream>>>(vals, rows, cols, x1, x2, 2.0f);
  }
  // Stage 4: WMMA GEMM + bias (12500 waves, 8 waves per 256-thread block)
  {
    int threads = 256;
    int tiles = (B_BATCH * V_NODES) / 16;               // 12500
    int blocks = (tiles * 32 + threads - 1) / threads;  // 1563
    cheb_gemm<<<blocks, threads, 0, stream>>>(x0, x1, x2, weight, bias, out);
  }
}